// NNLoss_6459630813839
// MI455X (gfx1250) — compile-verified
//
#include <hip/hip_runtime.h>

// Bidirectional 2-D NN (Chamfer) loss for B=64, N=M=2048, F=4, D=2.
//
// d2[n,m] = |p|^2 + |t|^2 - 2 p.t  is computed 16x16 at a time with
// V_WMMA_F32_16X16X4_F32 by packing A rows = [px, py, |p|^2, 1] and
// B cols = [-2tx, -2ty, 1, |t|^2] (K=4 rank-2 trick), so one WMMA yields a
// full 16x16 distance tile. VALU only does min/argmin bookkeeping.

typedef __attribute__((ext_vector_type(2))) float v2f;
typedef __attribute__((ext_vector_type(8))) float v8f;

#define BATCH 64
#define NPTS 2048
#define TILES (NPTS / 16)                 // 128
#define WAVES_PER_DIR (BATCH * TILES)     // 8192
#define WAVES_PER_BLOCK 8                 // 256 threads = 8 wave32

__device__ __forceinline__ float wave_sum32(float x) {
#pragma unroll
  for (int m = 16; m >= 1; m >>= 1) x += __shfl_xor(x, m, 32);
  return x;
}

// ---- Direction 1: for each pred row n, argmin over targ cols m -------------
__global__ void __launch_bounds__(256) nn_row_kernel(
    const float4* __restrict__ preds, const float4* __restrict__ targs,
    const float* __restrict__ subcoef, float* __restrict__ partial) {
  const int lane = threadIdx.x & 31;
  const int wg   = blockIdx.x * WAVES_PER_BLOCK + (threadIdx.x >> 5);
  const int b    = wg / TILES;
  const int n0   = (wg % TILES) * 16;
  const bool hi  = lane >= 16;
  const int l15  = lane & 15;

  // A operand (16x4 f32): lanes 0-15 hold K0/K1, lanes 16-31 hold K2/K3.
  const float4 p = preds[b * NPTS + n0 + l15];
  const float pp = p.x * p.x + p.y * p.y;
  v2f a;
  a.x = hi ? pp   : p.x;
  a.y = hi ? 1.0f : p.y;

  float mv[8];
  int   mi[8];
#pragma unroll
  for (int v = 0; v < 8; ++v) { mv[v] = 3.4e38f; mi[v] = 0; }

  for (int t = 0; t < TILES; ++t) {
    const int m0 = t * 16;
    const float4 q = targs[b * NPTS + m0 + l15];
    const float tt = q.x * q.x + q.y * q.y;
    // B operand (4x16 f32): VGPR0 = rows K0/K2, VGPR1 = rows K1/K3.
    v2f bb;
    bb.x = hi ? 1.0f : -2.0f * q.x;
    bb.y = hi ? tt   : -2.0f * q.y;

    v8f c = {};
    c = __builtin_amdgcn_wmma_f32_16x16x4_f32(
        /*neg_a=*/false, a, /*neg_b=*/false, bb,
        /*c_mod=*/(short)0, c, /*reuse_a=*/false, /*reuse_b=*/false);

    const int col = m0 + l15;  // this lane's column index
#pragma unroll
    for (int v = 0; v < 8; ++v) {   // VGPR v holds row (v or v+8), col = lane
      const float d = c[v];
      const bool better = d < mv[v];   // strict '<' keeps first (smallest m)
      mv[v] = better ? d   : mv[v];
      mi[v] = better ? col : mi[v];
    }
  }

  // Cross-lane argmin inside each 16-lane half (columns interleave by lane).
#pragma unroll
  for (int v = 0; v < 8; ++v) {
#pragma unroll
    for (int m = 8; m >= 1; m >>= 1) {
      const float ov = __shfl_xor(mv[v], m, 32);
      const int   oi = __shfl_xor(mi[v], m, 32);
      const bool take = (ov < mv[v]) || (ov == mv[v] && oi < mi[v]);
      mv[v] = take ? ov : mv[v];
      mi[v] = take ? oi : mi[v];
    }
  }

  float part = 0.0f;
  if (l15 == 0) {  // lane 0 -> rows n0..n0+7, lane 16 -> rows n0+8..n0+15
    const float sc0 = subcoef[0], sc1 = subcoef[1];
    const int rbase = hi ? 8 : 0;
#pragma unroll
    for (int v = 0; v < 8; ++v) {
      const float4 pr = preds[b * NPTS + n0 + rbase + v];
      const float4 tn = targs[b * NPTS + mi[v]];
      part += fabsf(pr.x - tn.x) * sc0 + fabsf(pr.y - tn.y) * sc1;
    }
  }
  part = wave_sum32(part);
  if (lane == 0) partial[wg] = part;
}

// ---- Direction 2: for each targ col m, argmin over pred rows n -------------
__global__ void __launch_bounds__(256) nn_col_kernel(
    const float4* __restrict__ preds, const float4* __restrict__ targs,
    float* __restrict__ partial) {
  const int lane = threadIdx.x & 31;
  const int wg   = blockIdx.x * WAVES_PER_BLOCK + (threadIdx.x >> 5);
  const int b    = wg / TILES;
  const int m0   = (wg % TILES) * 16;
  const bool hi  = lane >= 16;
  const int l15  = lane & 15;

  // B operand fixed for this wave: the 16 targ columns m0..m0+15.
  const float4 q = targs[b * NPTS + m0 + l15];
  const float tt = q.x * q.x + q.y * q.y;
  v2f bb;
  bb.x = hi ? 1.0f : -2.0f * q.x;
  bb.y = hi ? tt   : -2.0f * q.y;

  float mval = 3.4e38f;
  int   mrow = 0;

  for (int t = 0; t < TILES; ++t) {
    const int n0 = t * 16;
    const float4 p = preds[b * NPTS + n0 + l15];
    const float pp = p.x * p.x + p.y * p.y;
    v2f a;
    a.x = hi ? pp   : p.x;
    a.y = hi ? 1.0f : p.y;

    v8f c = {};
    c = __builtin_amdgcn_wmma_f32_16x16x4_f32(
        false, a, false, bb, (short)0, c, false, false);

    const int rb = n0 + (hi ? 8 : 0);  // rows this half covers, ascending in v
#pragma unroll
    for (int v = 0; v < 8; ++v) {
      const float d = c[v];
      const bool better = d < mval;    // rows visited in increasing order
      mval = better ? d        : mval;
      mrow = better ? (rb + v) : mrow;
    }
  }

  // Combine the lane pair (l, l+16): same column, disjoint row halves.
  {
    const float ov = __shfl_xor(mval, 16, 32);
    const int   oi = __shfl_xor(mrow, 16, 32);
    const bool take = (ov < mval) || (ov == mval && oi < mrow);
    mval = take ? ov : mval;
    mrow = take ? oi : mrow;
  }

  float part = 0.0f;
  if (!hi) {  // lanes 0-15 own columns m0..m0+15
    const float4 pn = preds[b * NPTS + mrow];
    part = fabsf(pn.x - q.x) + fabsf(pn.y - q.y);
  }
  part = wave_sum32(part);
  if (lane == 0) partial[WAVES_PER_DIR + wg] = part;
}

// ---- Deterministic final reduction (fixed tree order, no float atomics) ----
__global__ void __launch_bounds__(256) nn_reduce_kernel(
    const float* __restrict__ partial, float* __restrict__ out) {
  __shared__ float s[256];
  float acc = 0.0f;
  for (int i = threadIdx.x; i < 2 * WAVES_PER_DIR; i += 256) acc += partial[i];
  s[threadIdx.x] = acc;
  __syncthreads();
#pragma unroll
  for (int st = 128; st >= 1; st >>= 1) {
    if ((int)threadIdx.x < st) s[threadIdx.x] += s[threadIdx.x + st];
    __syncthreads();
  }
  if (threadIdx.x == 0) out[0] = s[0];
}

extern "C" void kernel_launch(void* const* d_in, const int* in_sizes, int n_in,
                              void* d_out, int out_size, void* d_ws, size_t ws_size,
                              hipStream_t stream) {
  (void)in_sizes; (void)n_in; (void)out_size; (void)ws_size;
  const float4* preds   = (const float4*)d_in[0];  // (64, 2048, 4) f32
  const float4* targs   = (const float4*)d_in[1];  // (64, 2048, 4) f32
  const float*  subcoef = (const float*)d_in[2];   // (2,) f32
  float* out     = (float*)d_out;                  // scalar
  float* partial = (float*)d_ws;                   // 2*8192 floats = 64 KB

  const dim3 block(256);
  const dim3 grid(WAVES_PER_DIR / WAVES_PER_BLOCK);  // 1024 blocks

  nn_row_kernel<<<grid, block, 0, stream>>>(preds, targs, subcoef, partial);
  nn_col_kernel<<<grid, block, 0, stream>>>(preds, targs, partial);
  nn_reduce_kernel<<<1, 256, 0, stream>>>(partial, out);
}